// GraphConv_4879082848619
// MI455X (gfx1250) — compile-verified
//
#include <hip/hip_runtime.h>

#define HID 256
#define IN_CH 512
#define OUT_CH 40
#define BN_EPS 1e-5f

typedef __attribute__((ext_vector_type(2))) float v2f;
typedef __attribute__((ext_vector_type(8))) float v8f;

// ---------------------------------------------------------------- utilities
__device__ __forceinline__ void atomAddF(float* p, float v) {
  // lowers to global_atomic_add_f32 (no pre-op return)
  __hip_atomic_fetch_add(p, v, __ATOMIC_RELAXED, __HIP_MEMORY_SCOPE_AGENT);
}

// CDNA5 async global->LDS copy, 16B per lane, tracked by ASYNCcnt.
// lds_off: wave-relative LDS byte offset (= low 32 bits of a generic
// pointer into __shared__ memory, per ISA flat-LDS addressing).
__device__ __forceinline__ void async_copy_b128(unsigned lds_off, const float* gsrc) {
  asm volatile("global_load_async_to_lds_b128 %0, %1, off"
               :: "v"(lds_off), "v"(gsrc) : "memory");
}
__device__ __forceinline__ void wait_async0() {
  asm volatile("s_wait_asynccnt 0x0" ::: "memory");
}
__device__ __forceinline__ unsigned lds_offset_of(const void* p) {
  return (unsigned)(unsigned long long)p;
}

__global__ void zero4_kernel(float4* __restrict__ p, int n4) {
  int i = blockIdx.x * blockDim.x + threadIdx.x;
  if (i < n4) p[i] = make_float4(0.f, 0.f, 0.f, 0.f);
}

// Per-column affine coefficients. stage s in {0,1,2}: y = gemm*scale + shift,
// scale = gamma*rsqrt(var+eps), shift = (bias - mean)*scale + beta.
// stage 3 (classifier): scale = 1, shift = clf_b (zero-padded to 256).
__global__ void coef_kernel(const float* __restrict__ fc0_b,
                            const float* __restrict__ conv_b,
                            const float* __restrict__ gamma,
                            const float* __restrict__ beta,
                            const float* __restrict__ mean,
                            const float* __restrict__ var,
                            const float* __restrict__ clf_b,
                            float* __restrict__ coef) {
  int c = threadIdx.x;  // 0..255
  for (int s = 0; s < 3; ++s) {
    float g  = gamma[s * HID + c];
    float bt = beta[s * HID + c];
    float mn = mean[s * HID + c];
    float vr = var[s * HID + c];
    float sc = g * rsqrtf(vr + BN_EPS);
    float bias = (s == 0) ? fc0_b[c] : conv_b[(s - 1) * HID + c];
    coef[s * 2 * HID + c]       = sc;
    coef[s * 2 * HID + HID + c] = (bias - mn) * sc + bt;
  }
  coef[3 * 2 * HID + c]       = 1.0f;
  coef[3 * 2 * HID + HID + c] = (c < OUT_CH) ? clf_b[c] : 0.0f;
}

// ---------------------------------------------------------------- degrees
__global__ void deg_count_kernel(const int* __restrict__ col, float* __restrict__ deg, int E) {
  int i = blockIdx.x * blockDim.x + threadIdx.x;
  if (i < E) atomAddF(&deg[col[i]], 1.0f);
}

__global__ void deg_rsqrt_kernel(float* __restrict__ deg, int n) {
  int i = blockIdx.x * blockDim.x + threadIdx.x;
  if (i < n) {
    float d = deg[i];
    deg[i] = (d > 0.0f) ? rsqrtf(d) : 0.0f;
  }
}

// ---------------------------------------------------------------- SpMM
// out[col[e], :] += dn[row[e]]*dn[col[e]] * h[row[e], :]   (HID = 256)
// One wave per edge (grid-stride). Lane j owns channels [8j, 8j+8): the wave's
// two b128 gathers cover one contiguous 1KB row; scatter is 8 consecutive
// global_atomic_add_f32 per lane. h/out are L2-resident (102MB vs 192MB L2).
__global__ void spmm_kernel(const float* __restrict__ h,
                            const int* __restrict__ row,
                            const int* __restrict__ col,
                            const float* __restrict__ dn,
                            float* __restrict__ out, int E) {
  int gtid = blockIdx.x * blockDim.x + threadIdx.x;
  int lane = gtid & 31;
  int wid  = gtid >> 5;
  int nw   = (gridDim.x * blockDim.x) >> 5;
  for (int e = wid; e < E; e += nw) {
    int r = row[e];
    int c = col[e];
    float v = dn[r] * dn[c];
    const float4* __restrict__ hr4 = (const float4*)(h + (size_t)r * HID) + lane * 2;
    float4 h0 = hr4[0];
    float4 h1 = hr4[1];
    float* __restrict__ oc = out + (size_t)c * HID + lane * 8;
    atomAddF(oc + 0, v * h0.x);
    atomAddF(oc + 1, v * h0.y);
    atomAddF(oc + 2, v * h0.z);
    atomAddF(oc + 3, v * h0.w);
    atomAddF(oc + 4, v * h1.x);
    atomAddF(oc + 5, v * h1.y);
    atomAddF(oc + 6, v * h1.z);
    atomAddF(oc + 7, v * h1.w);
  }
}

// ---------------------------------------------------------------- main GEMM
// 80 rows x 256 cols per block (100000 = 80*1250, no row guards).
// 16 waves; wave w owns columns [16w,16w+16); each wave runs 5 row-subtiles
// per B fetch (5x W-reuse, 5 WMMA per B load pair). The 80xK A panel is
// consumed in K-chunks of 128, double-buffered in LDS and staged with
// CDNA5 global_load_async_to_lds_b128 (ASYNCcnt) overlapping the WMMA work.
//
// VGPR layouts per CDNA5 ISA 7.12.2:
//   A 16x4 : lanes 0-15 hold (k0,k0+1) of row M=lane; lanes 16-31 (k0+2,k0+3)
//   B 4x16 : lanes 0-15 rows k0,k0+1 at N=lane; lanes 16-31 rows k0+2,k0+3
//   C/D    : VGPR v -> row v (lanes 0-15) / v+8 (lanes 16-31), N = lane&15
template <int K, bool RELU, bool RESID>
__global__ __launch_bounds__(512) void gemm80_wmma_kernel(
    const float* __restrict__ A, const float* __restrict__ W,
    const float* __restrict__ coef, const float* __restrict__ resid,
    float* __restrict__ out) {
  constexpr int KC = 128;         // K-chunk staged per buffer
  constexpr int KP = KC + 4;      // padded LDS row stride (banks step by 4)
  constexpr int NCHUNK = K / KC;
  constexpr int MR = 80;          // rows per block (5 subtiles of 16)
  __shared__ float As[2 * MR * KP];

  const int tid = threadIdx.x;
  const int wave = tid >> 5;
  const int lane = tid & 31;
  const int half = lane >> 4;
  const int l15  = lane & 15;
  const int col  = wave * 16 + l15;
  const int rowBase = blockIdx.x * MR;

  // Stage chunk c into buffer b: MR*KC floats = 2560 x b128, 5 per thread.
  auto stage = [&](int c, int b) {
    const float* src = A + (size_t)rowBase * K + c * KC;
    const unsigned ldsBase = lds_offset_of(&As[b * MR * KP]);
#pragma unroll
    for (int t = 0; t < 5; ++t) {
      int idx = (tid + t * 512) * 4;  // float index within MR x KC chunk
      int r = idx / KC;
      int cc = idx - r * KC;
      async_copy_b128(ldsBase + 4u * (unsigned)(r * KP + cc),
                      src + (size_t)r * K + cc);
    }
  };

  stage(0, 0);
  wait_async0();
  __syncthreads();

  v8f acc[5];
#pragma unroll
  for (int s = 0; s < 5; ++s) acc[s] = (v8f){0.f, 0.f, 0.f, 0.f, 0.f, 0.f, 0.f, 0.f};

  for (int c = 0; c < NCHUNK; ++c) {
    const int cur = c & 1;
    if (c + 1 < NCHUNK) stage(c + 1, cur ^ 1);  // overlap next stage with math
    const float* __restrict__ as = &As[cur * MR * KP];
    for (int k0 = 0; k0 < KC; k0 += 4) {
      const int kk = k0 + half * 2;
      const int kg = c * KC + kk;
      v2f b;
      b.x = W[(size_t)kg * HID + col];
      b.y = W[(size_t)(kg + 1) * HID + col];
#pragma unroll
      for (int s = 0; s < 5; ++s) {
        const float* ap = as + (l15 + 16 * s) * KP + kk;
        v2f a;
        a.x = ap[0];
        a.y = ap[1];
        acc[s] = __builtin_amdgcn_wmma_f32_16x16x4_f32(false, a, false, b,
                                                       (short)0, acc[s], false, false);
      }
    }
    wait_async0();
    __syncthreads();
  }

  // fused epilogue: y = acc*scale + shift ; relu ; + residual ; store
  const float sc = coef[col];
  const float sh = coef[HID + col];
#pragma unroll
  for (int s = 0; s < 5; ++s) {
#pragma unroll
    for (int v = 0; v < 8; ++v) {
      int m = rowBase + s * 16 + v + half * 8;
      float val = acc[s][v] * sc + sh;
      if (RELU) val = fmaxf(val, 0.0f);
      if (RESID) val += resid[(size_t)m * HID + col];
      out[(size_t)m * HID + col] = val;
    }
  }
}

// ---------------------------------------------------------------- classifier
// 16 rows x 40 cols per block (3 padded 16-wide tiles).
// D column n depends only on B data held by lanes n / n+16, so lanes with
// col >= 40 may compute garbage columns that are never stored: W column index
// is clamped (in-bounds loads, no mask, no EXEC manipulation in the K loop);
// only the stores are guarded.
__global__ __launch_bounds__(96) void clf_wmma_kernel(
    const float* __restrict__ A, const float* __restrict__ W,
    const float* __restrict__ coef, float* __restrict__ out) {
  constexpr int K = HID;
  constexpr int KP = K + 4;
  __shared__ float As[16 * KP];

  const int tid = threadIdx.x;
  const int rowBase = blockIdx.x * 16;

  for (int idx = tid; idx < 16 * K; idx += 96) {
    int r = idx / K;
    int c = idx % K;
    As[r * KP + c] = A[(size_t)(rowBase + r) * K + c];
  }
  __syncthreads();

  const int wave = tid >> 5;
  const int lane = tid & 31;
  const int half = lane >> 4;
  const int l15  = lane & 15;
  const int col  = wave * 16 + l15;                    // < 48
  const int colW = (col < OUT_CH) ? col : (OUT_CH - 1);  // clamp, hoisted

  v8f acc = {0.f, 0.f, 0.f, 0.f, 0.f, 0.f, 0.f, 0.f};
  const float* __restrict__ as = &As[l15 * KP];
  const float* __restrict__ wp = W + colW;
  for (int k0 = 0; k0 < K; k0 += 4) {
    int kk = k0 + half * 2;
    v2f a;
    a.x = as[kk];
    a.y = as[kk + 1];
    v2f b;
    b.x = wp[(size_t)kk * OUT_CH];
    b.y = wp[(size_t)(kk + 1) * OUT_CH];
    acc = __builtin_amdgcn_wmma_f32_16x16x4_f32(false, a, false, b, (short)0,
                                                acc, false, false);
  }

  float sh = coef[HID + col];  // scale is 1 for the classifier stage
#pragma unroll
  for (int v = 0; v < 8; ++v) {
    int m = rowBase + v + half * 8;
    float val = acc[v] + sh;
    if (col < OUT_CH) out[(size_t)m * OUT_CH + col] = val;
  }
}

// ---------------------------------------------------------------- launcher
extern "C" void kernel_launch(void* const* d_in, const int* in_sizes, int n_in,
                              void* d_out, int out_size, void* d_ws, size_t ws_size,
                              hipStream_t stream) {
  const float* x      = (const float*)d_in[0];
  const int*   ei     = (const int*)d_in[1];
  const float* fc0_W  = (const float*)d_in[2];
  const float* fc0_b  = (const float*)d_in[3];
  const float* conv_W = (const float*)d_in[4];
  const float* conv_b = (const float*)d_in[5];
  const float* gamma  = (const float*)d_in[6];
  const float* beta   = (const float*)d_in[7];
  const float* mean   = (const float*)d_in[8];
  const float* var    = (const float*)d_in[9];
  const float* clf_W  = (const float*)d_in[10];
  const float* clf_b  = (const float*)d_in[11];
  float* out = (float*)d_out;

  const int Nn = in_sizes[0] / IN_CH;  // 100000 (divisible by 80 and 16)
  const int E  = in_sizes[1] / 2;      // 3200000
  const int* row = ei;
  const int* col = ei + E;

  float* h0   = (float*)d_ws;                // [N,256] first-layer features (residual)
  float* hcur = h0 + (size_t)Nn * HID;       // [N,256] current features
  float* agg  = hcur + (size_t)Nn * HID;     // [N,256] SpMM accumulator
  float* dn   = agg + (size_t)Nn * HID;      // [N] 1/sqrt(degree)
  float* coef = dn + Nn;                     // 4 stages x (scale[256], shift[256])

  const int nh4 = Nn * HID / 4;
  const int blocks80 = Nn / 80;   // 1250
  const int blocks16 = Nn / 16;   // 6250

  coef_kernel<<<1, 256, 0, stream>>>(fc0_b, conv_b, gamma, beta, mean, var, clf_b, coef);

  // degree normalization
  zero4_kernel<<<(Nn / 4 + 255) / 256, 256, 0, stream>>>((float4*)dn, Nn / 4);
  deg_count_kernel<<<(E + 255) / 256, 256, 0, stream>>>(col, dn, E);
  deg_rsqrt_kernel<<<(Nn + 255) / 256, 256, 0, stream>>>(dn, Nn);

  // h0 = relu(bn0(x @ fc0_W + b))
  gemm80_wmma_kernel<IN_CH, true, false>
      <<<blocks80, 512, 0, stream>>>(x, fc0_W, coef + 0 * 2 * HID, nullptr, h0);

  // layer 0: agg = Anorm @ h0 ; hcur = relu(bn1(agg @ W0 + b0)) + h0
  zero4_kernel<<<(nh4 + 255) / 256, 256, 0, stream>>>((float4*)agg, nh4);
  spmm_kernel<<<4096, 256, 0, stream>>>(h0, row, col, dn, agg, E);
  gemm80_wmma_kernel<HID, true, true>
      <<<blocks80, 512, 0, stream>>>(agg, conv_W, coef + 1 * 2 * HID, h0, hcur);

  // layer 1: agg = Anorm @ hcur ; hcur = relu(bn2(agg @ W1 + b1)) + h0
  zero4_kernel<<<(nh4 + 255) / 256, 256, 0, stream>>>((float4*)agg, nh4);
  spmm_kernel<<<4096, 256, 0, stream>>>(hcur, row, col, dn, agg, E);
  gemm80_wmma_kernel<HID, true, true>
      <<<blocks80, 512, 0, stream>>>(agg, conv_W + HID * HID, coef + 2 * 2 * HID, h0, hcur);

  // out = hcur @ clf_W + clf_b
  clf_wmma_kernel<<<blocks16, 96, 0, stream>>>(hcur, clf_W, coef + 3 * 2 * HID, out);
}